// MambaBlock_8418135900520
// MI455X (gfx1250) — compile-verified
//
#include <hip/hip_runtime.h>
#include <hip/hip_bf16.h>
#include <math.h>

typedef __attribute__((ext_vector_type(2))) float v2f;
typedef __attribute__((ext_vector_type(4))) float v4f;
typedef __attribute__((ext_vector_type(8))) float v8f;

#define D_MODEL_C 1024
#define D_STATE_C 16
#define D_CONV_C  4
#define DT_RANK_C 64
#define D_INNER_C 2048
#define BATCH_C   2
#define SEQ_C     1024
#define NTOK_C    (BATCH_C * SEQ_C)   // 2048 tokens

// ---------------------------------------------------------------------------
// Generic fp32 WMMA GEMM:  C[M,N] = A[M,lda(first K cols)] * W[N,K]^T
// One wave computes a 16(M) x 64(N) strip: 4 accumulators reuse each A frag.
// Branch-free WMMA region: N-tail tiles are pointer-clamped (compute garbage,
// skip store) so EXEC is never touched around v_wmma. K must be % 16 == 0.
// EPILOG: 0 = plain store, 1 = softplus(acc + bias[n])
// f32 16x16x4 WMMA VGPR layout (ISA 7.12.2):
//   A (16x4):  lane<16 -> row=lane, a[0]=K, a[1]=K+1 ; lane>=16 -> K+2,K+3
//   B (4x16):  lane<16 -> col=lane, b[0]=row K, b[1]=row K+1 ; lane>=16 -> K+2,K+3
//   C (16x16): VGPR i: lanes 0-15 row m0+i, lanes 16-31 row m0+i+8
// ---------------------------------------------------------------------------
template <int EPILOG>
__global__ __launch_bounds__(256)
void wmma_gemm_f32(const float* __restrict__ A,
                   const float* __restrict__ W,
                   const float* __restrict__ bias,
                   float* __restrict__ C,
                   int M, int N, int K, int lda)
{
    const int lane      = threadIdx.x & 31;
    const int waveInBlk = threadIdx.x >> 5;
    const int nStrips   = (N + 63) >> 6;                 // 64-wide N strips
    const int strip     = blockIdx.x * 8 + waveInBlk;    // 8 waves / block
    const int total     = (M >> 4) * nStrips;
    if (strip >= total) return;

    const int m0 = (strip / nStrips) << 4;
    const int n0 = (strip % nStrips) << 6;

    const int lhalf = lane >> 4;   // 0 or 1 (selects K pair)
    const int lidx  = lane & 15;   // A row / B,C column within tile

    v8f acc[4];
    acc[0] = v8f{}; acc[1] = v8f{}; acc[2] = v8f{}; acc[3] = v8f{};

    // Per-lane base pointers (contiguous float pairs -> b64 loads).
    // Invalid N-tail tiles are clamped to column 0: harmless garbage math,
    // never stored -> no divergence in the WMMA loop.
    const float* arow = A + (size_t)(m0 + lidx) * lda + 2 * lhalf;
    const float* bcol[4];
#pragma unroll
    for (int t = 0; t < 4; ++t) {
        int nc = n0 + 16 * t;
        bcol[t] = W + (size_t)((nc < N ? nc : 0) + lidx) * K + 2 * lhalf;
    }

    for (int k = 0; k < K; k += 16) {
        v2f a[4];
        v2f b[4][4];
        // ---- batch all loads first: ~20 b64 loads in flight ----
#pragma unroll
        for (int kk = 0; kk < 4; ++kk) {
            int kb = k + 4 * kk;
            a[kk][0] = arow[kb];
            a[kk][1] = arow[kb + 1];
#pragma unroll
            for (int t = 0; t < 4; ++t) {
                b[kk][t][0] = bcol[t][kb];
                b[kk][t][1] = bcol[t][kb + 1];
            }
        }
        // ---- then 16 back-to-back WMMAs ----
#pragma unroll
        for (int kk = 0; kk < 4; ++kk) {
#pragma unroll
            for (int t = 0; t < 4; ++t) {
                acc[t] = __builtin_amdgcn_wmma_f32_16x16x4_f32(
                    false, a[kk], false, b[kk][t], (short)0, acc[t],
                    false, false);
            }
        }
    }

#pragma unroll
    for (int t = 0; t < 4; ++t) {
        int nc = n0 + 16 * t;
        if (nc >= N) break;                               // store-side only
        float* cp = C + (size_t)(m0 + 8 * lhalf) * N + nc + lidx;
        float bv = (EPILOG == 1) ? bias[nc + lidx] : 0.f;
#pragma unroll
        for (int i = 0; i < 8; ++i) {
            float v = acc[t][i];
            if (EPILOG == 1) {
                float x = v + bv;                         // softplus
                v = (x > 20.f) ? x : log1pf(__expf(x));
            }
            cp[(size_t)i * N] = v;
        }
    }
}

// ---------------------------------------------------------------------------
// Depthwise causal conv1d (width 4, left pad 3) + bias + SiLU.
// x_in = xz[:, 0:2048] of row-major [NTOK, 4096]; out x_act [NTOK, 2048].
// ---------------------------------------------------------------------------
__global__ __launch_bounds__(256)
void conv_silu_kernel(const float* __restrict__ xz,
                      const float* __restrict__ conv_w,
                      const float* __restrict__ conv_b,
                      float* __restrict__ x_act)
{
    int idx = blockIdx.x * blockDim.x + threadIdx.x;  // over NTOK*D_INNER
    if (idx >= NTOK_C * D_INNER_C) return;
    int d = idx & (D_INNER_C - 1);
    int m = idx >> 11;            // token index b*SEQ + t
    int t = m & (SEQ_C - 1);

    float acc = conv_b[d];
    const float* w = conv_w + d * D_CONV_C;
#pragma unroll
    for (int j = 0; j < D_CONV_C; ++j) {
        int tt = t - (D_CONV_C - 1) + j;
        if (tt >= 0)
            acc += w[j] * xz[(size_t)(m - (D_CONV_C - 1) + j) * (2 * D_INNER_C) + d];
    }
    float s = acc / (1.f + __expf(-acc));              // SiLU
    x_act[(size_t)m * D_INNER_C + d] = s;
}

// ---------------------------------------------------------------------------
// Selective scan: one thread per (batch, channel). State h[16] in VGPRs.
// B_t/C_t fetched as float4 (b128 loads, wave-broadcast addresses).
// Fuses +u*D and *silu(z). Writes y in place over the dt buffer.
// ---------------------------------------------------------------------------
__global__ __launch_bounds__(256)
void selective_scan_kernel(const float* __restrict__ xz,     // [NTOK,4096] (z cols 2048..4095)
                           const float* __restrict__ x_act,  // [NTOK,2048] (u)
                           const float* __restrict__ dbc,    // [NTOK,96] (B 64..79, C 80..95)
                           float* __restrict__ dt_y,         // in: dt [NTOK,2048]; out: y
                           const float* __restrict__ A_log,  // [2048,16]
                           const float* __restrict__ Dvec)   // [2048]
{
    int gid = blockIdx.x * blockDim.x + threadIdx.x;   // 0..4095
    if (gid >= BATCH_C * D_INNER_C) return;
    int b = gid >> 11;
    int d = gid & (D_INNER_C - 1);

    float Arow[D_STATE_C];
#pragma unroll
    for (int s = 0; s < D_STATE_C; ++s)
        Arow[s] = -__expf(A_log[(size_t)d * D_STATE_C + s]);
    const float Dd = Dvec[d];

    float h[D_STATE_C];
#pragma unroll
    for (int s = 0; s < D_STATE_C; ++s) h[s] = 0.f;

    for (int t = 0; t < SEQ_C; ++t) {
        size_t m = (size_t)b * SEQ_C + t;
        float u   = x_act[m * D_INNER_C + d];
        float dtv = dt_y [m * D_INNER_C + d];
        // 96-float rows, 16B-aligned at offsets 64 and 80 -> aligned b128 loads
        const v4f* bc4 = (const v4f*)(dbc + m * 96 + 64);
        v4f Bv0 = bc4[0], Bv1 = bc4[1], Bv2 = bc4[2], Bv3 = bc4[3];
        float Bt[D_STATE_C], Ct[D_STATE_C];
#pragma unroll
        for (int q = 0; q < 4; ++q) {
            Bt[q]      = Bv0[q]; Bt[4 + q]  = Bv1[q];
            Bt[8 + q]  = Bv2[q]; Bt[12 + q] = Bv3[q];
        }
        const v4f* cc4 = (const v4f*)(dbc + m * 96 + 80);
        v4f Cv0 = cc4[0], Cv1 = cc4[1], Cv2 = cc4[2], Cv3 = cc4[3];
#pragma unroll
        for (int q = 0; q < 4; ++q) {
            Ct[q]      = Cv0[q]; Ct[4 + q]  = Cv1[q];
            Ct[8 + q]  = Cv2[q]; Ct[12 + q] = Cv3[q];
        }

        float du = dtv * u;
        float y = 0.f;
#pragma unroll
        for (int s = 0; s < D_STATE_C; ++s) {
            float dA = __expf(dtv * Arow[s]);          // v_exp_f32 (TRANS)
            h[s] = dA * h[s] + du * Bt[s];
            y += h[s] * Ct[s];
        }
        y += u * Dd;
        float z  = xz[m * (2 * D_INNER_C) + D_INNER_C + d];
        float sz = z / (1.f + __expf(-z));             // silu(z)
        dt_y[m * D_INNER_C + d] = y * sz;
    }
}

// ---------------------------------------------------------------------------
static inline int gemm_blocks(int M, int N)
{
    int strips = (M >> 4) * ((N + 63) >> 6);
    return (strips + 7) >> 3;                          // 8 waves per 256-thread block
}

extern "C" void kernel_launch(void* const* d_in, const int* in_sizes, int n_in,
                              void* d_out, int out_size, void* d_ws, size_t ws_size,
                              hipStream_t stream) {
    (void)in_sizes; (void)n_in; (void)out_size; (void)ws_size;

    const float* x      = (const float*)d_in[0];  // [2,1024,1024]
    const float* W_in   = (const float*)d_in[1];  // [4096,1024]
    const float* conv_w = (const float*)d_in[2];  // [2048,1,4]
    const float* conv_b = (const float*)d_in[3];  // [2048]
    const float* W_x    = (const float*)d_in[4];  // [96,2048]
    const float* W_dt   = (const float*)d_in[5];  // [2048,64]
    const float* b_dt   = (const float*)d_in[6];  // [2048]
    const float* A_log  = (const float*)d_in[7];  // [2048,16]
    const float* Dvec   = (const float*)d_in[8];  // [2048]
    const float* W_out  = (const float*)d_in[9];  // [1024,2048]
    float* out = (float*)d_out;                   // [2,1024,1024]

    // Workspace layout (floats): xz | x_act | dbc | dt(->y in place)
    float* xz   = (float*)d_ws;                              // 2048*4096
    float* xact = xz   + (size_t)NTOK_C * (2 * D_INNER_C);   // 2048*2048
    float* dbc  = xact + (size_t)NTOK_C * D_INNER_C;         // 2048*96
    float* dtb  = dbc  + (size_t)NTOK_C * 96;                // 2048*2048

    // 1) in_proj: xz = x @ W_in^T      [2048,1024]x[4096,1024]^T
    wmma_gemm_f32<0><<<gemm_blocks(NTOK_C, 2 * D_INNER_C), 256, 0, stream>>>(
        x, W_in, nullptr, xz, NTOK_C, 2 * D_INNER_C, D_MODEL_C, D_MODEL_C);

    // 2) depthwise causal conv + SiLU -> x_act
    {
        int n = NTOK_C * D_INNER_C;
        conv_silu_kernel<<<(n + 255) / 256, 256, 0, stream>>>(xz, conv_w, conv_b, xact);
    }

    // 3) x_proj: dbc = x_act @ W_x^T   [2048,2048]x[96,2048]^T
    wmma_gemm_f32<0><<<gemm_blocks(NTOK_C, 96), 256, 0, stream>>>(
        xact, W_x, nullptr, dbc, NTOK_C, 96, D_INNER_C, D_INNER_C);

    // 4) dt = softplus(dt_lr @ W_dt^T + b_dt); dt_lr = dbc[:, :64] (lda=96)
    wmma_gemm_f32<1><<<gemm_blocks(NTOK_C, D_INNER_C), 256, 0, stream>>>(
        dbc, W_dt, b_dt, dtb, NTOK_C, D_INNER_C, DT_RANK_C, 96);

    // 5) selective scan (+u*D, *silu(z)); y overwrites dt buffer
    selective_scan_kernel<<<(BATCH_C * D_INNER_C) / 256, 256, 0, stream>>>(
        xz, xact, dbc, dtb, A_log, Dvec);

    // 6) out_proj: out = y @ W_out^T   [2048,2048]x[1024,2048]^T
    wmma_gemm_f32<0><<<gemm_blocks(NTOK_C, D_MODEL_C), 256, 0, stream>>>(
        dtb, W_out, nullptr, out, NTOK_C, D_MODEL_C, D_INNER_C, D_INNER_C);
}